// ComponentTree_32762010534549
// MI455X (gfx1250) — compile-verified
//
#include <hip/hip_runtime.h>

#define MT_H 64
#define MT_W 64
#define MT_N (MT_H * MT_W)
#define MT_C 3
#define MT_BLOCK 1024

// CDNA5 async global->LDS path (ASYNCcnt-tracked). The builtin exists on this
// toolchain (probe passed in round 1); its params are addrspace-qualified
// int* ("__device__ int *" == addrspace(1) in clang's HIP diagnostics).
#if defined(__gfx1250__) && __has_builtin(__builtin_amdgcn_global_load_async_to_lds_b32)
#define MT_ASYNC_LDS 1
typedef __attribute__((address_space(1))) int mt_gint;  // global (AS1)
typedef __attribute__((address_space(3))) int mt_lint;  // LDS    (AS3)
#endif

__global__ __launch_bounds__(MT_BLOCK) void maxtree_kernel(
    const float* __restrict__ vw, float* __restrict__ out) {
  const int c = blockIdx.x;     // channel: one workgroup per channel
  const int tid = threadIdx.x;  // 0..1023 (32 waves of 32)

  __shared__ float f[MT_N];            // 16 KB: channel altitudes
  __shared__ int order[MT_N];          // 16 KB: pixels in decreasing altitude
  __shared__ int parent[MT_N];         // 16 KB
  __shared__ int zpar[MT_N];           // 16 KB: union-find forest
  __shared__ unsigned char proc[MT_N]; //  4 KB  (total 68 KB of the 320 KB WGP LDS)

  // ---- Phase 0: gather channel c (stride-3 float gather) into LDS ----------
#ifdef MT_ASYNC_LDS
  for (int i = tid; i < MT_N; i += MT_BLOCK) {
    __builtin_amdgcn_global_load_async_to_lds_b32(
        (mt_gint*)(vw + i * MT_C + c), (mt_lint*)(&f[i]), 0, 0);
    proc[i] = 0;
  }
#if __has_builtin(__builtin_amdgcn_s_wait_asynccnt)
  __builtin_amdgcn_s_wait_asynccnt(0);
#else
  asm volatile("s_wait_asynccnt 0" ::: "memory");
#endif
#else
  __builtin_prefetch(vw + tid * MT_C + c, 0, 1);  // global_prefetch_b8
  for (int i = tid; i < MT_N; i += MT_BLOCK) {
    f[i] = vw[i * MT_C + c];
    proc[i] = 0;
  }
#endif
  __syncthreads();

  // ---- Phase 1 (parallel): stable descending rank == argsort(-f) ----------
  // rank[i] = #{j : f[j] > f[i]}  +  #{j : f[j] == f[i] && j < i}
  // All ranks are distinct, so order[] is a permutation; LDS reads of f[j]
  // are uniform across lanes per iteration (bank-broadcast friendly).
  for (int i = tid; i < MT_N; i += MT_BLOCK) {
    const float v = f[i];
    int r = 0;
    for (int j = 0; j < MT_N; ++j) {
      const float fj = f[j];
      r += (int)(fj > v) + (int)((fj == v) & (j < i));
    }
    order[r] = i;
  }
  __syncthreads();

  // ---- Phases 2+3 (serial, loop-carried dependence): lane 0 in LDS --------
  if (tid == 0) {
    // Berger union-find sweep in decreasing altitude
    for (int i = 0; i < MT_N; ++i) {
      const int p = order[i];
      parent[p] = p;
      zpar[p] = p;
      proc[p] = 1;
      const int row = p >> 6;
      const int col = p & (MT_W - 1);
      int nb[4], ok[4];
      nb[0] = p - MT_W; ok[0] = (row > 0);
      nb[1] = p + MT_W; ok[1] = (row < MT_H - 1);
      nb[2] = p - 1;    ok[2] = (col > 0);
      nb[3] = p + 1;    ok[3] = (col < MT_W - 1);
      for (int j = 0; j < 4; ++j) {
        if (!ok[j]) continue;
        const int n = nb[j];
        if (!proc[n]) continue;
        // find root (with path compression; root-equivalent to reference)
        int r = n;
        while (zpar[r] != r) r = zpar[r];
        int x = n;
        while (zpar[x] != r) { const int nx = zpar[x]; zpar[x] = r; x = nx; }
        if (r != p) {
          parent[r] = p;
          zpar[r] = p;
        }
      }
    }
    // canonicalization: root -> leaves == reverse processing order
    for (int i = MT_N - 1; i >= 0; --i) {
      const int p = order[i];
      const int q = parent[p];
      const int pq = parent[q];
      parent[p] = (f[pq] == f[q]) ? pq : q;
    }
  }
  __syncthreads();

  // ---- Phase 4 (parallel): write outputs ----------------------------------
  // d_out layout (float32): parents [C, N] as values, then altitudes [C*N].
  for (int i = tid; i < MT_N; i += MT_BLOCK) {
    out[c * MT_N + i] = (float)parent[i];
    out[MT_C * MT_N + c * MT_N + i] = f[i];
  }
}

extern "C" void kernel_launch(void* const* d_in, const int* in_sizes, int n_in,
                              void* d_out, int out_size, void* d_ws, size_t ws_size,
                              hipStream_t stream) {
  (void)in_sizes; (void)n_in; (void)out_size; (void)d_ws; (void)ws_size;
  const float* vw = (const float*)d_in[0];  // [64,64,3] float32, HWC
  float* out = (float*)d_out;               // 24576 floats
  maxtree_kernel<<<MT_C, MT_BLOCK, 0, stream>>>(vw, out);
}